// Struc2Vec_90091234001037
// MI455X (gfx1250) — compile-verified
//
#include <hip/hip_runtime.h>
#include <hip/hip_bf16.h>

typedef __attribute__((ext_vector_type(2))) float v2f;
typedef __attribute__((ext_vector_type(8))) float v8f;

#define Vn 20000
#define Pn 40000
#define Dn 40000
#define Nn 100000
#define En 1250000
#define Bn 1024

// lrelu(x) = where(x>=0, x, 0.01x) == max(x, 0.01x) since 0 < slope < 1.
// Lowers to v_mul_f32 + v_max_f32 instead of cmp/cndmask chains.
__device__ __forceinline__ float lrelu_f(float v) {
    return fmaxf(v, 0.01f * v);
}

// ---------------------------------------------------------------------------
// zero buffer (float4)
// ---------------------------------------------------------------------------
__global__ void zero_f4(float4* __restrict__ p, int n4) {
    int t = blockIdx.x * blockDim.x + threadIdx.x;
    if (t < n4) p[t] = make_float4(0.f, 0.f, 0.f, 0.f);
}

// ---------------------------------------------------------------------------
// Edge precompute: agg_ti[dst[e]] += lrelu(ea[e]*W3 + b3)   (16 threads/edge)
// ---------------------------------------------------------------------------
__global__ void edge_theta_scatter(const float* __restrict__ ea,
                                   const int* __restrict__ ei,
                                   const float* __restrict__ W3,
                                   const float* __restrict__ b3,
                                   float* __restrict__ agg) {
    int t = blockIdx.x * blockDim.x + threadIdx.x;
    int e = t >> 4;
    if (e >= En) return;
    int q = t & 15;                 // handles cols 4q..4q+3
    int d = ei[En + e];             // edge_index[1][e]
    float a = ea[e];
    float* dp = agg + (size_t)d * 64 + q * 4;
#pragma unroll
    for (int u = 0; u < 4; ++u) {
        int j = q * 4 + u;
        float v = lrelu_f(fmaf(a, W3[j], b3[j]));
        atomicAdd(dp + u, v);
    }
}

// ---------------------------------------------------------------------------
// Gather/scatter: agg[dst[e]] += mu[src[e]]   (16 threads/edge, float4)
// ---------------------------------------------------------------------------
__global__ void mu_scatter(const float* __restrict__ mu,
                           const int* __restrict__ ei,
                           float* __restrict__ agg) {
    int t = blockIdx.x * blockDim.x + threadIdx.x;
    int e = t >> 4;
    if (e >= En) return;
    int q = t & 15;
    int s = ei[e];
    int d = ei[En + e];
    float4 v = ((const float4*)(mu + (size_t)s * 64))[q];
    float* dp = agg + (size_t)d * 64 + q * 4;
    atomicAdd(dp + 0, v.x);
    atomicAdd(dp + 1, v.y);
    atomicAdd(dp + 2, v.z);
    atomicAdd(dp + 3, v.w);
}

// ---------------------------------------------------------------------------
// inv += type_add + b1  (node types are contiguous blocks 0|1|2)
// ---------------------------------------------------------------------------
__global__ void add_type_and_b1(float* __restrict__ inv,
                                const float* __restrict__ xv,
                                const float* __restrict__ xp,
                                const float* __restrict__ xd,
                                const float* __restrict__ Wt0, const float* __restrict__ bt0,
                                const float* __restrict__ Wt1, const float* __restrict__ bt1,
                                const float* __restrict__ Wt2, const float* __restrict__ bt2,
                                const float* __restrict__ b1) {
    int t = blockIdx.x * blockDim.x + threadIdx.x;
    int i = t >> 6;
    if (i >= Nn) return;
    int j = t & 63;
    float v;
    if (i < Vn) {
        const float* x = xv + (size_t)i * 2;
        v = fmaf(x[0], Wt0[j], fmaf(x[1], Wt0[64 + j], bt0[j]));
    } else if (i < Vn + Pn) {
        const float* x = xp + (size_t)(i - Vn) * 3;
        v = fmaf(x[0], Wt1[j], fmaf(x[1], Wt1[64 + j], fmaf(x[2], Wt1[128 + j], bt1[j])));
    } else {
        const float* x = xd + (size_t)(i - Vn - Pn) * 2;
        v = fmaf(x[0], Wt2[j], fmaf(x[1], Wt2[64 + j], bt2[j]));
    }
    inv[(size_t)i * 64 + j] += v + b1[j];
}

// ---------------------------------------------------------------------------
// WMMA fp32 GEMM: out[M x 64] = act( in[M x 64] @ W[64 x 64] + bias + extra )
// One wave per 16-row band; 4 col-tiles of 16; K in steps of 4 via
// v_wmma_f32_16x16x4_f32. EXEC all-1 for every active wave (M % 16 == 0).
// A layout (ISA 7.12.2): M = lane%16; lanes 0-15 hold K={k0,k0+1},
// lanes 16-31 hold K={k0+2,k0+3}. B mirrored with N = lane%16.
// C/D: VGPR r -> M = r + 8*(lane>=16), N = lane%16.
// ---------------------------------------------------------------------------
__launch_bounds__(256)
__global__ void gemm64_wmma(const float* __restrict__ in,
                            const float* __restrict__ W,
                            const float* __restrict__ bias,
                            const float* __restrict__ extra,
                            float* __restrict__ out,
                            int Mrows, int act) {
    int wave = blockIdx.x * (blockDim.x >> 5) + (threadIdx.x >> 5);
    int lane = threadIdx.x & 31;
    int m0 = wave * 16;
    if (m0 >= Mrows) return;        // wave-uniform guard
    int lm = lane & 15;
    int hi = lane >> 4;

    // Preload the A fragments for the whole 16x64 band (reused by 4 col tiles)
    v2f afr[16];
    const float* arow = in + (size_t)(m0 + lm) * 64 + 2 * hi;
#pragma unroll
    for (int kk = 0; kk < 16; ++kk) {
        afr[kk][0] = arow[kk * 4 + 0];
        afr[kk][1] = arow[kk * 4 + 1];
    }

    for (int nt = 0; nt < 4; ++nt) {
        int col = nt * 16 + lm;
        float bv = bias ? bias[col] : 0.0f;
        v8f c;
#pragma unroll
        for (int r = 0; r < 8; ++r) {
            float cv = bv;
            if (extra) cv += extra[(size_t)(m0 + hi * 8 + r) * 64 + col];
            c[r] = cv;
        }
#pragma unroll
        for (int kk = 0; kk < 16; ++kk) {
            v2f b;
            b[0] = W[(size_t)(kk * 4 + 2 * hi + 0) * 64 + col];
            b[1] = W[(size_t)(kk * 4 + 2 * hi + 1) * 64 + col];
            c = __builtin_amdgcn_wmma_f32_16x16x4_f32(
                    false, afr[kk], false, b, (short)0, c, false, false);
        }
#pragma unroll
        for (int r = 0; r < 8; ++r) {
            float v = c[r];
            if (act) v = lrelu_f(v);
            out[(size_t)(m0 + hi * 8 + r) * 64 + col] = v;
        }
    }
}

// ---------------------------------------------------------------------------
// Batch scatter-mean accumulation (16 threads/node)
// ---------------------------------------------------------------------------
__global__ void batch_scatter(const float* __restrict__ mu,
                              const int* __restrict__ batch,
                              float* __restrict__ sums,
                              float* __restrict__ counts) {
    int t = blockIdx.x * blockDim.x + threadIdx.x;
    int i = t >> 4;
    if (i >= Nn) return;
    int q = t & 15;
    int b = batch[i];
    float4 v = ((const float4*)(mu + (size_t)i * 64))[q];
    float* dp = sums + (size_t)b * 64 + q * 4;
    atomicAdd(dp + 0, v.x);
    atomicAdd(dp + 1, v.y);
    atomicAdd(dp + 2, v.z);
    atomicAdd(dp + 3, v.w);
    if (q == 0) atomicAdd(&counts[b], 1.0f);
}

__global__ void normalize_emb(const float* __restrict__ sums,
                              const float* __restrict__ counts,
                              float* __restrict__ emb) {
    int t = blockIdx.x * blockDim.x + threadIdx.x;
    if (t >= Bn * 64) return;
    int b = t >> 6;
    float cnt = fmaxf(counts[b], 1.0f);
    emb[t] = sums[t] / cnt;
}

// ---------------------------------------------------------------------------
// Head: out[b] = sigmoid( dot(h[b], Wc2) + bc2 )   (one wave per graph)
// ---------------------------------------------------------------------------
__global__ void head_kernel(const float* __restrict__ h,
                            const float* __restrict__ Wc2,
                            const float* __restrict__ bc2,
                            float* __restrict__ out) {
    int g = blockIdx.x * (blockDim.x >> 5) + (threadIdx.x >> 5);
    if (g >= Bn) return;
    int lane = threadIdx.x & 31;
    const float* hr = h + (size_t)g * 64;
    float s = fmaf(hr[lane], Wc2[lane], hr[lane + 32] * Wc2[lane + 32]);
#pragma unroll
    for (int off = 16; off > 0; off >>= 1) s += __shfl_down(s, off, 32);
    if (lane == 0) {
        float z = s + bc2[0];
        out[g] = 1.0f / (1.0f + expf(-z));
    }
}

// ---------------------------------------------------------------------------
extern "C" void kernel_launch(void* const* d_in, const int* in_sizes, int n_in,
                              void* d_out, int out_size, void* d_ws, size_t ws_size,
                              hipStream_t stream) {
    const float* xv  = (const float*)d_in[0];
    const float* xp  = (const float*)d_in[1];
    const float* xd  = (const float*)d_in[2];
    const float* ea  = (const float*)d_in[3];
    const float* mu0 = (const float*)d_in[4];
    const float* W1  = (const float*)d_in[5];
    const float* b1  = (const float*)d_in[6];
    const float* W2  = (const float*)d_in[7];
    const float* b2  = (const float*)d_in[8];
    const float* W3  = (const float*)d_in[9];
    const float* b3  = (const float*)d_in[10];
    const float* Wt0 = (const float*)d_in[11];
    const float* bt0 = (const float*)d_in[12];
    const float* Wt1 = (const float*)d_in[13];
    const float* bt1 = (const float*)d_in[14];
    const float* Wt2 = (const float*)d_in[15];
    const float* bt2 = (const float*)d_in[16];
    const float* Wc1 = (const float*)d_in[17];
    const float* bc1 = (const float*)d_in[18];
    const float* Wc2 = (const float*)d_in[19];
    const float* bc2 = (const float*)d_in[20];
    const int*   ei    = (const int*)d_in[21];
    const int*   batch = (const int*)d_in[23];
    float* out = (float*)d_out;

    // workspace layout (floats)
    float* ws = (float*)d_ws;
    const size_t N64 = (size_t)Nn * 64;
    float* inv    = ws;                      // N x 64 : theta2_part + type_add + b1
    float* agg    = ws + N64;                // N x 64 : scatter accumulator
    float* mub    = ws + 2 * N64;            // N x 64 : mu working buffer
    float* sums   = ws + 3 * N64;            // B x 64
    float* counts = sums + (size_t)Bn * 64;  // B
    float* emb    = counts + Bn;             // B x 64
    float* hbuf   = emb + (size_t)Bn * 64;   // B x 64

    const int TB = 256;
    const int zeroBlk   = (int)((N64 / 4 + TB - 1) / TB);
    const int edgeBlk   = (En * 16 + TB - 1) / TB;
    const int gemmNBlk  = ((Nn / 16) + 7) / 8;   // 8 waves per block
    const int gemmBBlk  = ((Bn / 16) + 7) / 8;
    const int typeBlk   = (Nn * 64 + TB - 1) / TB;
    const int nodeBlk   = (Nn * 16 + TB - 1) / TB;

    // --- loop-invariant precompute -----------------------------------------
    zero_f4<<<zeroBlk, TB, 0, stream>>>((float4*)agg, (int)(N64 / 4));
    edge_theta_scatter<<<edgeBlk, TB, 0, stream>>>(ea, ei, W3, b3, agg);
    gemm64_wmma<<<gemmNBlk, TB, 0, stream>>>(agg, W2, b2, nullptr, inv, Nn, 0);
    add_type_and_b1<<<typeBlk, TB, 0, stream>>>(inv, xv, xp, xd,
                                                Wt0, bt0, Wt1, bt1, Wt2, bt2, b1);

    // --- R = 4 message-passing iterations ----------------------------------
    const float* cur = mu0;
    for (int it = 0; it < 4; ++it) {
        zero_f4<<<zeroBlk, TB, 0, stream>>>((float4*)agg, (int)(N64 / 4));
        mu_scatter<<<edgeBlk, TB, 0, stream>>>(cur, ei, agg);
        gemm64_wmma<<<gemmNBlk, TB, 0, stream>>>(agg, W1, nullptr, inv, mub, Nn, 1);
        cur = mub;
    }

    // --- graph readout + classifier ----------------------------------------
    zero_f4<<<(int)((Bn * 64 / 4 + TB - 1) / TB), TB, 0, stream>>>((float4*)sums, Bn * 64 / 4);
    zero_f4<<<1, TB, 0, stream>>>((float4*)counts, Bn / 4);
    batch_scatter<<<nodeBlk, TB, 0, stream>>>(cur, batch, sums, counts);
    normalize_emb<<<(Bn * 64 + TB - 1) / TB, TB, 0, stream>>>(sums, counts, emb);
    gemm64_wmma<<<gemmBBlk, TB, 0, stream>>>(emb, Wc1, bc1, nullptr, hbuf, Bn, 0);
    head_kernel<<<(Bn + 7) / 8, TB, 0, stream>>>(hbuf, Wc2, bc2, out);
}